// TNTLoss_42030549958864
// MI455X (gfx1250) — compile-verified
//
#include <hip/hip_runtime.h>
#include <hip/hip_bf16.h>

// ---------------------------------------------------------------------------
// TNT loss, MI455X (gfx1250, wave32).
// Memory-bound streaming reduction: ~92 MB @ 23.3 TB/s ~ 4 us floor.
//  - tnt_main: b128 streaming loads (2 elements/thread/step), scalar-hoisted
//    per-block lens/base (block always lies in one batch row).
//  - tnt_final: 16 chained V_WMMA_F32_16X16X4_F32 (A = ones) fold 1024 f32
//    partials with full f32 accumulation, deterministically.
// ---------------------------------------------------------------------------

#define BATCH 512
#define NCAND 8192
#define NBLOCKS 1024          // main kernel blocks; must stay 1024 (16 WMMAs * 64)
#define THREADS 256
#define PAIRS_PER_STEP (THREADS * 2)   // 512 elements per step
#define STEPS 8                        // 1024 blocks * 4096 elems == 512*8192

typedef float v2f __attribute__((ext_vector_type(2)));
typedef float v8f __attribute__((ext_vector_type(8)));

// ---- Kernel 1: prefix scans over B=512 + yaw smooth-L1 --------------------
__global__ void tnt_scan_yaw(const int* __restrict__ lens,
                             const int* __restrict__ ttc,
                             const float* __restrict__ y_yaw,
                             const float* __restrict__ yaw_pred,
                             int* __restrict__ base_out,
                             float* __restrict__ yaw_out) {
    __shared__ int s_l[BATCH];
    __shared__ int s_t[BATCH];
    __shared__ float s_f[16];
    const int t = threadIdx.x;          // 0..511
    const int myl = lens[t];
    const int myt = ttc[t] - 7;         // ttc_eff in [1,31)
    s_l[t] = myl;
    s_t[t] = myt;
    __syncthreads();
    // Hillis-Steele inclusive scan (9 steps)
    for (int off = 1; off < BATCH; off <<= 1) {
        int a = (t >= off) ? s_l[t - off] : 0;
        int b = (t >= off) ? s_t[t - off] : 0;
        __syncthreads();
        s_l[t] += a;
        s_t[t] += b;
        __syncthreads();
    }
    base_out[t] = s_l[t] - myl;         // exclusive cumsum of lens
    // yaw smooth-L1 contribution
    const int yidx = s_t[t] - 1;        // inclusive cumsum - 1, always >= 0
    const float gt_yaw = y_yaw[yidx] * 0.017453292519943295f + 3.14159265358979323846f;
    const float d = fabsf(yaw_pred[t] - gt_yaw);
    float sl = (d < 1.0f) ? (0.5f * d * d) : (d - 0.5f);
    // reduce 512 -> 1 (wave32 butterflies + LDS)
    for (int o = 16; o > 0; o >>= 1) sl += __shfl_xor(sl, o, 32);
    if ((t & 31) == 0) s_f[t >> 5] = sl;
    __syncthreads();
    if (t < 16) {
        float v = s_f[t];
        for (int o = 8; o > 0; o >>= 1) v += __shfl_xor(v, o, 16);
        if (t == 0) yaw_out[0] = v;
    }
}

// ---- Kernel 2: streaming BCE + masked-L1, one partial per block -----------
// Each block owns 4096 consecutive elements of one batch row b = blockIdx.x/2,
// so lens[b]/base[b] are uniform (scalar loads). Each thread handles 2
// adjacent elements per step -> b128 loads on rcnn_cls/offset_pred, b64 on gt.
__global__ void tnt_main(const float* __restrict__ offset_pred,  // (B,N,2)
                         const float* __restrict__ rcnn_cls,     // (B,N,2)
                         const int* __restrict__ gt,             // (B,N)
                         const float* __restrict__ offset_each,  // (B*N,2)
                         const int* __restrict__ lens,           // (B,)
                         const int* __restrict__ base,           // (B,)
                         float* __restrict__ partials) {
    const int t = threadIdx.x;
    const int b = blockIdx.x >> 1;                 // uniform: scalar loads below
    const int len_b = lens[b];
    const int base_b = base[b];
    const int n0 = (blockIdx.x & 1) * 4096 + t * 2;  // n of first elem of pair
    const int e0 = b * NCAND + n0;                   // global element index
    float bce_acc = 0.0f;
    float off_acc = 0.0f;
#pragma unroll
    for (int i = 0; i < STEPS; ++i) {
        const int e = e0 + i * PAIRS_PER_STEP;       // even
        const int n = n0 + i * PAIRS_PER_STEP;
        const float4 rc = ((const float4*)rcnn_cls)[e >> 1];     // elems e, e+1
        const float4 op = ((const float4*)offset_pred)[e >> 1];
        const int2 lab2 = ((const int2*)gt)[e >> 1];
        // ---------- element e (rc.x, rc.y) ----------
        {
            const float x = rc.y;
            const float lab = (float)lab2.x;
            const float tl = log1pf(expf(-fabsf(x)));
            const float ls_pos = fminf(x, 0.0f) - tl;    // log_sigmoid(x)
            const float ls_neg = fminf(-x, 0.0f) - tl;   // log_sigmoid(-x)
            bce_acc += -(10.0f * lab * ls_pos + (1.0f - lab) * ls_neg);
            float g0 = 0.0f, g1 = 0.0f;
            if ((rc.y > rc.x) && (n < len_b)) {          // argmax==1 && valid
                int idx = base_b + n;
                const int maxi = BATCH * NCAND - 1;
                idx = (idx > maxi) ? maxi : idx;
                const float2 g = ((const float2*)offset_each)[idx];
                g0 = g.x; g1 = g.y;
            }
            off_acc += fabsf(op.x - g0) + fabsf(op.y - g1);
        }
        // ---------- element e+1 (rc.z, rc.w) ----------
        {
            const float x = rc.w;
            const float lab = (float)lab2.y;
            const float tl = log1pf(expf(-fabsf(x)));
            const float ls_pos = fminf(x, 0.0f) - tl;
            const float ls_neg = fminf(-x, 0.0f) - tl;
            bce_acc += -(10.0f * lab * ls_pos + (1.0f - lab) * ls_neg);
            float g0 = 0.0f, g1 = 0.0f;
            if ((rc.w > rc.z) && ((n + 1) < len_b)) {
                int idx = base_b + n + 1;
                const int maxi = BATCH * NCAND - 1;
                idx = (idx > maxi) ? maxi : idx;
                const float2 g = ((const float2*)offset_each)[idx];
                g0 = g.x; g1 = g.y;
            }
            off_acc += fabsf(op.z - g0) + fabsf(op.w - g1);
        }
    }
    // wave32 butterfly reduce
    for (int o = 16; o > 0; o >>= 1) {
        bce_acc += __shfl_xor(bce_acc, o, 32);
        off_acc += __shfl_xor(off_acc, o, 32);
    }
    __shared__ float s_b[8];
    __shared__ float s_o[8];
    if ((t & 31) == 0) {
        s_b[t >> 5] = bce_acc;
        s_o[t >> 5] = off_acc;
    }
    __syncthreads();
    if (t == 0) {
        float sb = 0.0f, so = 0.0f;
#pragma unroll
        for (int w = 0; w < 8; ++w) { sb += s_b[w]; so += s_o[w]; }
        partials[blockIdx.x] = 0.002f * sb + 0.004f * so;   // RCNN_CLS_W, OFFSET_W
    }
}

// ---- Kernel 3: WMMA reduction of 1024 partials + yaw -> scalar ------------
// One wave32, EXEC all-ones. A = ones(16x4): D[m][n] = sum_k B[k][n] + C[m][n].
// Every row of D is the identical column-sum vector, so the 32-lane butterfly
// of c[0] yields exactly 2x the grand total (lane L and L+16 hold identical
// values; doubling is fp-exact), hence *0.5f.
__global__ void tnt_final(const float* __restrict__ partials,
                          const float* __restrict__ yaw,
                          float* __restrict__ out) {
    const int l = threadIdx.x;   // 0..31
    v2f a;
    a.x = 1.0f;
    a.y = 1.0f;
    v8f c = {};
#pragma unroll
    for (int g = 0; g < 16; ++g) {
        v2f bm;
        bm.x = partials[g * 64 + l];
        bm.y = partials[g * 64 + 32 + l];
        c = __builtin_amdgcn_wmma_f32_16x16x4_f32(
                /*neg_a=*/false, a, /*neg_b=*/false, bm,
                /*c_mod=*/(short)0, c, /*reuse_a=*/false, /*reuse_b=*/false);
    }
    float v = c[0];
    for (int o = 16; o > 0; o >>= 1) v += __shfl_xor(v, o, 32);
    if (l == 0) out[0] = v * 0.5f + yaw[0];
}

// ---------------------------------------------------------------------------
extern "C" void kernel_launch(void* const* d_in, const int* in_sizes, int n_in,
                              void* d_out, int out_size, void* d_ws, size_t ws_size,
                              hipStream_t stream) {
    // setup_inputs order (d_in[0] target_prob_pred is unused by the reference)
    const float* offset_pred     = (const float*)d_in[1];
    const float* rcnn_cls        = (const float*)d_in[2];
    const float* target_pred_yaw = (const float*)d_in[3];
    const int*   gt_target_prob  = (const int*)d_in[4];
    const float* offset_each     = (const float*)d_in[5];
    const int*   lens            = (const int*)d_in[6];
    const int*   ttc             = (const int*)d_in[7];
    const float* y_yaw           = (const float*)d_in[8];

    // workspace layout
    int*   ws_base = (int*)d_ws;                          // 512 * 4B
    float* ws_yaw  = (float*)((char*)d_ws + 2048);        // 1 float
    float* ws_part = (float*)((char*)d_ws + 2560);        // 1024 floats

    tnt_scan_yaw<<<1, BATCH, 0, stream>>>(lens, ttc, y_yaw, target_pred_yaw,
                                          ws_base, ws_yaw);
    tnt_main<<<NBLOCKS, THREADS, 0, stream>>>(offset_pred, rcnn_cls, gt_target_prob,
                                              offset_each, lens, ws_base, ws_part);
    tnt_final<<<1, 32, 0, stream>>>(ws_part, ws_yaw, (float*)d_out);
}